// BiGN_64287070486723
// MI455X (gfx1250) — compile-verified
//
#include <hip/hip_runtime.h>
#include <hip/hip_bf16.h>

#define N_USERS 100000
#define N_ITEMS 50000
#define N_TOT   (N_USERS + N_ITEMS)
#define DIM     64
#define N_EDGE  2400000
#define LAYERS  3
#define BATCH   4096

typedef __attribute__((ext_vector_type(2))) float v2f;
typedef __attribute__((ext_vector_type(8))) float v8f;

// ---------------------------------------------------------------------------
// all_emb = concat(user_emb, item_emb); emb_sum = all_emb (layer-0 term)
// ---------------------------------------------------------------------------
__global__ void init_emb_kernel(const float* __restrict__ user_emb,
                                const float* __restrict__ item_emb,
                                float* __restrict__ all_emb,
                                float* __restrict__ emb_sum) {
    long long i = (long long)blockIdx.x * blockDim.x + threadIdx.x;   // float4 idx
    const long long tot4  = (long long)N_TOT * DIM / 4;
    const long long user4 = (long long)N_USERS * DIM / 4;
    if (i >= tot4) return;
    float4 v = (i < user4) ? ((const float4*)user_emb)[i]
                           : ((const float4*)item_emb)[i - user4];
    ((float4*)all_emb)[i] = v;
    ((float4*)emb_sum)[i] = v;
}

// ---------------------------------------------------------------------------
// Zero the two scatter accumulators
// ---------------------------------------------------------------------------
__global__ void zero2_kernel(float* __restrict__ a, float* __restrict__ b) {
    long long i = (long long)blockIdx.x * blockDim.x + threadIdx.x;   // float4 idx
    const long long tot4 = (long long)N_TOT * DIM / 4;
    if (i >= tot4) return;
    float4 z = make_float4(0.f, 0.f, 0.f, 0.f);
    ((float4*)a)[i] = z;
    ((float4*)b)[i] = z;
}

// ---------------------------------------------------------------------------
// SpMM scatter: acc[dst] += val * emb[src].  16 lanes per edge, float4 each.
// Accumulator is L2-resident (38.4 MB << 192 MB); f32 atomics execute at L2.
// ---------------------------------------------------------------------------
__global__ void spmm_scatter_kernel(const int* __restrict__ src,
                                    const int* __restrict__ dst,
                                    const float* __restrict__ val,
                                    const float* __restrict__ emb,
                                    float* __restrict__ acc) {
    long long gid = (long long)blockIdx.x * blockDim.x + threadIdx.x;
    long long e = gid >> 4;
    if (e >= (long long)N_EDGE) return;
    int part = (int)(gid & 15);
    int s = src[e];
    int d = dst[e];
    float w = val[e];
    float4 x = *(const float4*)(emb + (long long)s * DIM + part * 4);
    float* o = acc + (long long)d * DIM + part * 4;
    atomicAdd(o + 0, w * x.x);
    atomicAdd(o + 1, w * x.y);
    atomicAdd(o + 2, w * x.z);
    atomicAdd(o + 3, w * x.w);
}

// ---------------------------------------------------------------------------
// Attention blend: one wave32 per node row.
//   att_side = exp(mean_d(side*all + side)) = exp(sum_d side_d*(all_d+1)/64)
//   all_emb  = w*side + (1-w)*sim ;  emb_sum += all_emb
// ---------------------------------------------------------------------------
__global__ void attn_combine_kernel(const float* __restrict__ side,
                                    const float* __restrict__ simm,
                                    float* __restrict__ all_emb,
                                    float* __restrict__ emb_sum) {
    int lane = threadIdx.x & 31;
    int wave = threadIdx.x >> 5;
    long long n = (long long)blockIdx.x * (blockDim.x >> 5) + wave;
    if (n >= (long long)N_TOT) return;

    long long off = n * DIM + lane * 2;
    v2f s = *(const v2f*)(side + off);
    v2f m = *(const v2f*)(simm + off);
    v2f e = *(const v2f*)(all_emb + off);

    float ps = s.x * (e.x + 1.0f) + s.y * (e.y + 1.0f);
    float pm = m.x * (e.x + 1.0f) + m.y * (e.y + 1.0f);
    #pragma unroll
    for (int dlt = 16; dlt > 0; dlt >>= 1) {
        ps += __shfl_xor(ps, dlt, 32);
        pm += __shfl_xor(pm, dlt, 32);
    }
    float as = expf(ps * (1.0f / 64.0f));
    float am = expf(pm * (1.0f / 64.0f));
    float w = as / (as + am);

    v2f nw;
    nw.x = w * s.x + (1.0f - w) * m.x;
    nw.y = w * s.y + (1.0f - w) * m.y;
    *(v2f*)(all_emb + off) = nw;

    v2f acc = *(v2f*)(emb_sum + off);
    acc.x += nw.x;
    acc.y += nw.y;
    *(v2f*)(emb_sum + off) = acc;
}

// ---------------------------------------------------------------------------
// Final batched dot via V_WMMA_F32_16X16X4_F32.
// One wave handles 16 (user,item) pairs: A = user rows (16x4 slice),
// B = item rows (4x16 slice), 16 accumulating WMMAs cover D=64.
// out[pair i] = D[i][i] * (1/16)   (light_out = emb_sum/4 on both sides).
// A layout (ISA): lane L holds row L%16, K = (L/16)*2 + {0,1} -> contiguous
// float2 per lane; B mirrors with columns = item rows. EXEC all-ones before
// the divergent diagonal writeback, as WMMA requires.
// ---------------------------------------------------------------------------
__global__ void final_dot_wmma_kernel(const float* __restrict__ emb_sum,
                                      const int* __restrict__ users,
                                      const int* __restrict__ items,
                                      float* __restrict__ out) {
    int lane = threadIdx.x & 31;
    int wave = threadIdx.x >> 5;
    int base = (blockIdx.x * (blockDim.x >> 5) + wave) * 16;
    if (base >= BATCH) return;

    int r = lane & 15;
    int khalf = (lane >> 4) << 1;                 // 0 for lanes 0-15, 2 for 16-31
    int up = users[base + r];
    int ip = items[base + r] + N_USERS;
    const float* urow = emb_sum + (long long)up * DIM;
    const float* irow = emb_sum + (long long)ip * DIM;

    v8f c = {};
    #pragma unroll
    for (int kc = 0; kc < 16; ++kc) {
        int k = kc * 4 + khalf;
        v2f a = *(const v2f*)(urow + k);
        v2f b = *(const v2f*)(irow + k);
        // (neg_a, A, neg_b, B, c_mod, C, reuse_a, reuse_b)
        c = __builtin_amdgcn_wmma_f32_16x16x4_f32(
                false, a, false, b, (short)0, c, false, false);
    }

    // Diagonal extraction per 16x16 f32 C/D layout:
    //  i<8:  D[i][i] at lane i,  VGPR i
    //  i>=8: D[i][i] at lane i+16, VGPR i-8
    bool wr = false;
    int p = 0, idx = 0;
    if (lane < 8)        { wr = true; p = lane;      idx = lane;      }
    else if (lane >= 24) { wr = true; p = lane - 16; idx = lane - 24; }
    if (wr) {
        float dv;
        switch (idx) {
            case 0: dv = c[0]; break; case 1: dv = c[1]; break;
            case 2: dv = c[2]; break; case 3: dv = c[3]; break;
            case 4: dv = c[4]; break; case 5: dv = c[5]; break;
            case 6: dv = c[6]; break; default: dv = c[7]; break;
        }
        out[base + p] = dv * 0.0625f;   // (1/4)*(1/4)
    }
}

// ---------------------------------------------------------------------------
extern "C" void kernel_launch(void* const* d_in, const int* in_sizes, int n_in,
                              void* d_out, int out_size, void* d_ws, size_t ws_size,
                              hipStream_t stream) {
    (void)in_sizes; (void)n_in; (void)out_size; (void)ws_size;

    const float* user_emb  = (const float*)d_in[0];
    const float* item_emb  = (const float*)d_in[1];
    const int*   graph_src = (const int*)  d_in[2];
    const int*   graph_dst = (const int*)  d_in[3];
    const float* graph_val = (const float*)d_in[4];
    const int*   sim_src   = (const int*)  d_in[5];
    const int*   sim_dst   = (const int*)  d_in[6];
    const float* sim_val   = (const float*)d_in[7];
    const int*   users     = (const int*)  d_in[8];
    const int*   items     = (const int*)  d_in[9];
    float* out = (float*)d_out;

    // Workspace layout: 4 x (N_TOT*DIM) f32 = 4 x 38.4 MB = 153.6 MB
    size_t nd = (size_t)N_TOT * DIM;
    float* all_emb = (float*)d_ws;
    float* emb_sum = all_emb + nd;
    float* side    = emb_sum + nd;
    float* simm    = side    + nd;

    const long long tot4 = (long long)N_TOT * DIM / 4;
    const int blk = 256;

    init_emb_kernel<<<(unsigned)((tot4 + blk - 1) / blk), blk, 0, stream>>>(
        user_emb, item_emb, all_emb, emb_sum);

    const long long scatter_threads = (long long)N_EDGE * 16;
    for (int l = 0; l < LAYERS; ++l) {
        zero2_kernel<<<(unsigned)((tot4 + blk - 1) / blk), blk, 0, stream>>>(side, simm);
        spmm_scatter_kernel<<<(unsigned)((scatter_threads + blk - 1) / blk), blk, 0, stream>>>(
            graph_src, graph_dst, graph_val, all_emb, side);
        spmm_scatter_kernel<<<(unsigned)((scatter_threads + blk - 1) / blk), blk, 0, stream>>>(
            sim_src, sim_dst, sim_val, all_emb, simm);
        // 8 waves per block, one wave per node
        attn_combine_kernel<<<(unsigned)((N_TOT + 7) / 8), blk, 0, stream>>>(
            side, simm, all_emb, emb_sum);
    }

    // 8 waves/block * 16 pairs/wave = 128 pairs per block -> 32 blocks
    final_dot_wmma_kernel<<<BATCH / 128, blk, 0, stream>>>(emb_sum, users, items, out);
}